// Patch_Easre_66606352827267
// MI455X (gfx1250) — compile-verified
//
#include <hip/hip_runtime.h>

typedef __attribute__((ext_vector_type(2))) float v2f;
typedef __attribute__((ext_vector_type(4))) float v4f;
typedef __attribute__((ext_vector_type(8))) float v8f;

#define BATCH 2
#define CHN 256
#define HW 512
#define PS 52
#define NH 9
#define NP 81            // 9*9 patches
#define PS2 2704         // 52*52
#define COV 468          // 9*52 covered region
#define PLANE 262144     // 512*512
#define NCHUNK 16        // channel chunks per patch in pass 2
#define CPC 16           // channels per chunk

// Wave32 cross-lane sum using V_WMMA_F32_16X16X4_F32 with an all-ones B.
// A (16x4 f32) layout: a.x -> lanes 0-15: A[m,0], lanes 16-31: A[m,2];
// a.y = 0 zeroes columns 1,3. D = A*ones: every row of D is p[m]+p[m+16].
// Per-lane sum of the 8 D VGPRs gives half-sums; shfl_xor(16) completes.
__device__ __forceinline__ float waveReduceWmma(float p) {
  v2f a; a.x = p;    a.y = 0.0f;
  v2f o; o.x = 1.0f; o.y = 1.0f;
  v8f c = {};
  v8f d = __builtin_amdgcn_wmma_f32_16x16x4_f32(false, a, false, o, (short)0, c,
                                                false, false);
  float s = d[0] + d[1] + d[2] + d[3] + d[4] + d[5] + d[6] + d[7];
  s += __shfl_xor(s, 16, 32);
  return s;  // valid in all lanes
}

// 256-thread block reduction; result valid on thread 0 only.
__device__ __forceinline__ float blockReduce256(float v, float* red) {
  float ws = waveReduceWmma(v);
  int wid = threadIdx.x >> 5;
  if ((threadIdx.x & 31) == 0) red[wid] = ws;
  __syncthreads();
  float tot = 0.f;
  if (threadIdx.x == 0) {
#pragma unroll
    for (int i = 0; i < 8; ++i) tot += red[i];
  }
  return tot;
}

// Pass 1: g[b,c] = mean over the 512x512 plane. One block per (b,c).
__global__ void gap_kernel(const float* __restrict__ x, float* __restrict__ g) {
  __shared__ float red[8];
  int bc = blockIdx.x;
  const v4f* xv = (const v4f*)(x + (size_t)bc * PLANE);
  float s = 0.f;
#pragma unroll 4
  for (int i = threadIdx.x; i < PLANE / 4; i += 256) {
    v4f v = xv[i];
    s += (v.x + v.y) + (v.z + v.w);
  }
  float tot = blockReduce256(s, red);
  if (threadIdx.x == 0) g[bc] = tot * (1.0f / (float)PLANE);
}

// Pass 2: per (b, patch, channel-chunk) partial dot & sum-of-squares.
// Weight for flat element j = c*2704 + t is g[b, (c*144 + t) mod 256].
// Patch rows are 52 contiguous 16B-aligned floats -> 13 float4 per row.
__global__ void patch_stats_kernel(const float* __restrict__ x,
                                   const float* __restrict__ g,
                                   float* __restrict__ dotp,
                                   float* __restrict__ ssqp) {
  __shared__ float gs[CHN];
  __shared__ float redd[8];
  __shared__ float reds[8];
  int cc = blockIdx.x, p = blockIdx.y, b = blockIdx.z;
  int tid = threadIdx.x;
  gs[tid] = g[b * CHN + tid];
  __syncthreads();

  int ph = p / NH, pw = p - ph * NH;
  int c0 = cc * CPC;
  float dacc = 0.f, sacc = 0.f;
  for (int ci = 0; ci < CPC; ++ci) {
    int c = c0 + ci;
    const float* xp =
        x + (((size_t)(b * CHN + c)) * HW + (size_t)ph * PS) * HW + (size_t)pw * PS;
    int wbase = (c * 144) & 255;
    for (int t4 = tid; t4 < PS2 / 4; t4 += 256) {  // 676 float4s
      int kh = t4 / 13;
      int kw = (t4 - kh * 13) * 4;
      v4f v = *(const v4f*)(xp + kh * HW + kw);
      int wi = wbase + kh * PS + kw;
      dacc = fmaf(v.x, gs[wi & 255], dacc);
      dacc = fmaf(v.y, gs[(wi + 1) & 255], dacc);
      dacc = fmaf(v.z, gs[(wi + 2) & 255], dacc);
      dacc = fmaf(v.w, gs[(wi + 3) & 255], dacc);
      sacc = fmaf(v.x, v.x, sacc);
      sacc = fmaf(v.y, v.y, sacc);
      sacc = fmaf(v.z, v.z, sacc);
      sacc = fmaf(v.w, v.w, sacc);
    }
  }
  float dtot = blockReduce256(dacc, redd);
  float stot = blockReduce256(sacc, reds);
  if (tid == 0) {
    int o = (b * NP + p) * NCHUNK + cc;
    dotp[o] = dtot;
    ssqp[o] = stot;
  }
}

// Pass 3: finish cos, argmax (idx) / argmin (ids) with first-occurrence
// tie-break (matches jnp.argmax/argmin). One block per batch.
__global__ void select_kernel(const float* __restrict__ g,
                              const float* __restrict__ dotp,
                              const float* __restrict__ ssqp,
                              int* __restrict__ sel) {
  __shared__ float sh[128];
  __shared__ float cosv[NP];
  __shared__ float gnsh;
  int b = blockIdx.x, tid = threadIdx.x;

  float s = 0.f;
  for (int c = tid; c < CHN; c += 128) {
    float v = g[b * CHN + c];
    s += v * v;
  }
  sh[tid] = s;
  __syncthreads();
  for (int off = 64; off > 0; off >>= 1) {
    if (tid < off) sh[tid] += sh[tid + off];
    __syncthreads();
  }
  if (tid == 0) gnsh = sqrtf(sh[0]) * (float)PS;
  __syncthreads();

  if (tid < NP) {
    float d = 0.f, q = 0.f;
#pragma unroll
    for (int k = 0; k < NCHUNK; ++k) {
      d += dotp[(b * NP + tid) * NCHUNK + k];
      q += ssqp[(b * NP + tid) * NCHUNK + k];
    }
    cosv[tid] = d / fmaxf(gnsh * sqrtf(q), 1e-8f);
  }
  __syncthreads();

  if (tid == 0) {
    float bmax = cosv[0], bmin = cosv[0];
    int imax = 0, imin = 0;
    for (int i = 1; i < NP; ++i) {
      float cv = cosv[i];
      if (cv > bmax) { bmax = cv; imax = i; }
      if (cv < bmin) { bmin = cv; imin = i; }
    }
    sel[b] = imax;          // most similar patch (gets replaced)
    sel[BATCH + b] = imin;  // least similar patch (replacement source)
  }
}

// Pass 4: out = mask * x, float4 per thread.
// 512, 468 and 52 are all multiples of 4, so an aligned float4 never crosses
// a row, the covered/border boundary, or a patch boundary in w. The scramble
// gather j=(kh*52+kw)*256+c -> (j/2704, (j%2704)/52, j%52) differs per
// element, so the 4 mask loads stay scalar (L2-resident: each patch's 2.7MB
// source tile is reused across 256 channels). Out is streamed non-temporal.
__global__ void output_kernel(const float* __restrict__ x,
                              const int* __restrict__ sel,
                              float* __restrict__ out) {
  __shared__ int s_imax, s_imin;
  size_t blockBase = (size_t)blockIdx.x * 1024;  // 256 threads * 4 elems
  int b = (int)(blockBase >> 26);                // uniform per block
  if (threadIdx.x == 0) {
    s_imax = sel[b];
    s_imin = sel[BATCH + b];
  }
  __syncthreads();

  size_t lin = blockBase + (size_t)threadIdx.x * 4;
  int w0 = (int)(lin & (HW - 1));
  int h = (int)((lin >> 9) & (HW - 1));
  int c = (int)((lin >> 18) & (CHN - 1));

  v4f r = {0.f, 0.f, 0.f, 0.f};
  if (h < COV && w0 < COV) {
    int ph = h / PS, pw = w0 / PS;
    int kh = h - ph * PS, kw0 = w0 - pw * PS;
    int p = ph * NH + pw;
    int psrc = (p == s_imax) ? s_imin : p;
    int phs = psrc / NH, pws = psrc - phs * NH;

    int j0 = (kh * PS + kw0) * CHN + c;
    float m[4];
#pragma unroll
    for (int e = 0; e < 4; ++e) {
      int j = j0 + e * CHN;
      int cs = j / PS2;
      int rr = j - cs * PS2;
      int khs = rr / PS;
      int kws = rr - khs * PS;
      size_t src = (((size_t)(b * CHN + cs)) * HW + (size_t)(phs * PS + khs)) * HW +
                   (size_t)(pws * PS + kws);
      m[e] = x[src];
    }
    v4f xv = *(const v4f*)(x + lin);
    r.x = m[0] * xv.x;
    r.y = m[1] * xv.y;
    r.z = m[2] * xv.z;
    r.w = m[3] * xv.w;
  }
  __builtin_nontemporal_store(r, (v4f*)(out + lin));
}

extern "C" void kernel_launch(void* const* d_in, const int* in_sizes, int n_in,
                              void* d_out, int out_size, void* d_ws, size_t ws_size,
                              hipStream_t stream) {
  const float* x = (const float*)d_in[0];
  float* out = (float*)d_out;

  float* ws = (float*)d_ws;
  float* g = ws;                                 // 512 floats
  float* dotp = ws + 512;                        // 2*81*16 = 2592 floats
  float* ssqp = ws + 512 + BATCH * NP * NCHUNK;  // 2592 floats
  int* sel = (int*)(ws + 512 + 2 * BATCH * NP * NCHUNK);  // 4 ints

  gap_kernel<<<BATCH * CHN, 256, 0, stream>>>(x, g);

  dim3 grid2(NCHUNK, NP, BATCH);
  patch_stats_kernel<<<grid2, 256, 0, stream>>>(x, g, dotp, ssqp);

  select_kernel<<<BATCH, 128, 0, stream>>>(g, dotp, ssqp, sel);

  size_t total = (size_t)BATCH * CHN * HW * HW;
  output_kernel<<<(unsigned)(total / 1024), 256, 0, stream>>>(x, sel, out);
}